// GraphAttention_46162308497842
// MI455X (gfx1250) — compile-verified
//
#include <hip/hip_runtime.h>
#include <hip/hip_bf16.h>

typedef __attribute__((ext_vector_type(2))) float v2f;
typedef __attribute__((ext_vector_type(8))) float v8f;

#define IN_F   128
#define OUT_F  64
#define LDSW   132   // 128 + 4 pad floats: rows step 4 banks -> conflict-free b64 reads
#define NEG_SLOPE 0.01f
#define EPS 1e-6f

// ---------------------------------------------------------------------------
// Kernel 1: messages = x @ W.T via V_WMMA_F32_16X16X4_F32 (full fp32 precision)
// Block: 256 threads = 8 waves. Each wave computes a 16-row x 64-col strip
// (4 accumulators), reusing one A fragment across the 4 column tiles.
// W (64x128, 32KB) is staged in padded LDS once per block.
// ---------------------------------------------------------------------------
__global__ __launch_bounds__(256)
void gat_gemm(const float* __restrict__ x, const float* __restrict__ W,
              float* __restrict__ msgs, int N)
{
    __shared__ float Ws[OUT_F * LDSW];

    // Cooperative load of W into LDS (float4 chunks), row-padded to LDSW.
    for (int t = threadIdx.x; t < OUT_F * (IN_F / 4); t += 256) {
        int r  = t / (IN_F / 4);
        int c4 = t % (IN_F / 4);
        float4 v = reinterpret_cast<const float4*>(W)[r * (IN_F / 4) + c4];
        float* dst = &Ws[r * LDSW + c4 * 4];
        dst[0] = v.x; dst[1] = v.y; dst[2] = v.z; dst[3] = v.w;
    }
    __syncthreads();

    const int wave = threadIdx.x >> 5;
    const int lane = threadIdx.x & 31;
    const int lr   = lane & 15;   // row (A) / col (B) within tile
    const int hi   = lane >> 4;   // lane-group: selects K pair {0,1} vs {2,3}

    const int tile_row = blockIdx.x * 128 + wave * 16;
    int arow = tile_row + lr;
    if (arow >= N) arow = N - 1;            // branchless clamp: EXEC stays all-1s for WMMA
    const float* __restrict__ xrow = x + (size_t)arow * IN_F;

    v8f acc0 = {}, acc1 = {}, acc2 = {}, acc3 = {};

    #pragma unroll
    for (int k0 = 0; k0 < IN_F; k0 += 4) {
        // A fragment: 16x4 f32 layout -> lane holds (row=lr, K = k0 + 2*hi + {0,1})
        v2f afrag = *reinterpret_cast<const v2f*>(xrow + k0 + hi * 2);
        // B fragments: B[k][n] = W[n][k]; lane holds (col=c0+lr, K = k0 + 2*hi + {0,1})
        v2f b0 = *reinterpret_cast<const v2f*>(&Ws[( 0 + lr) * LDSW + k0 + hi * 2]);
        v2f b1 = *reinterpret_cast<const v2f*>(&Ws[(16 + lr) * LDSW + k0 + hi * 2]);
        v2f b2 = *reinterpret_cast<const v2f*>(&Ws[(32 + lr) * LDSW + k0 + hi * 2]);
        v2f b3 = *reinterpret_cast<const v2f*>(&Ws[(48 + lr) * LDSW + k0 + hi * 2]);

        acc0 = __builtin_amdgcn_wmma_f32_16x16x4_f32(false, afrag, false, b0, (short)0, acc0, false, false);
        acc1 = __builtin_amdgcn_wmma_f32_16x16x4_f32(false, afrag, false, b1, (short)0, acc1, false, false);
        acc2 = __builtin_amdgcn_wmma_f32_16x16x4_f32(false, afrag, false, b2, (short)0, acc2, false, false);
        acc3 = __builtin_amdgcn_wmma_f32_16x16x4_f32(false, afrag, false, b3, (short)0, acc3, false, false);
    }

    // C/D layout: VGPR i holds row (i + 8*hi), col = lr (per 16-col tile).
    #pragma unroll
    for (int i = 0; i < 8; ++i) {
        int row = tile_row + i + 8 * hi;
        if (row < N) {
            float* orow = msgs + (size_t)row * OUT_F;
            orow[ 0 + lr] = acc0[i];
            orow[16 + lr] = acc1[i];
            orow[32 + lr] = acc2[i];
            orow[48 + lr] = acc3[i];
        }
    }
}

// ---------------------------------------------------------------------------
// Kernel 2: per-node attention scalars alpha_dst = msg.a[:64], alpha_src = msg.a[64:]
// ---------------------------------------------------------------------------
__global__ __launch_bounds__(256)
void gat_alpha(const float* __restrict__ msgs, const float* __restrict__ a,
               float* __restrict__ alpha_dst, float* __restrict__ alpha_src, int N)
{
    int n = blockIdx.x * 256 + threadIdx.x;
    if (n >= N) return;
    const float4* m  = reinterpret_cast<const float4*>(msgs + (size_t)n * OUT_F);
    const float4* ad = reinterpret_cast<const float4*>(a);
    const float4* as = reinterpret_cast<const float4*>(a + OUT_F);
    float sd = 0.f, ss = 0.f;
    #pragma unroll
    for (int i = 0; i < OUT_F / 4; ++i) {
        float4 mv = m[i], dv = ad[i], sv = as[i];
        sd += mv.x * dv.x + mv.y * dv.y + mv.z * dv.z + mv.w * dv.w;
        ss += mv.x * sv.x + mv.y * sv.y + mv.z * sv.z + mv.w * sv.w;
    }
    alpha_dst[n] = sd;
    alpha_src[n] = ss;
}

// ---------------------------------------------------------------------------
// Kernel 3: self-loop term; also INITIALIZES out and denom (no memset needed).
// Self loop on node n: e = alpha_dst[n] + alpha_src[n]; w = exp(leaky(e)).
// ---------------------------------------------------------------------------
__global__ __launch_bounds__(256)
void gat_selfloop(const float* __restrict__ msgs,
                  const float* __restrict__ alpha_dst, const float* __restrict__ alpha_src,
                  float* __restrict__ out, float* __restrict__ denom, int N)
{
    int t = blockIdx.x * 256 + threadIdx.x;
    if (t >= N * OUT_F) return;
    int n = t >> 6;
    float e = alpha_dst[n] + alpha_src[n];
    e = (e >= 0.f) ? e : NEG_SLOPE * e;
    float w = __expf(e);
    out[t] = w * msgs[t];
    if ((t & 63) == 0) denom[n] = w;
}

// ---------------------------------------------------------------------------
// Kernel 4: edge scatter. One wave per edge (grid-stride). Each lane handles
// 2 contiguous features (float2 gather is fully coalesced: 256B/wave).
// agg lives in d_out and fits in L2 (25.6MB < 192MB) -> atomics stay on-chip.
// ---------------------------------------------------------------------------
__global__ __launch_bounds__(256)
void gat_edges(const int* __restrict__ ei, const float* __restrict__ msgs,
               const float* __restrict__ alpha_dst, const float* __restrict__ alpha_src,
               float* __restrict__ out, float* __restrict__ denom, int E)
{
    const int lane = threadIdx.x & 31;
    long long gw     = ((long long)blockIdx.x * blockDim.x + threadIdx.x) >> 5;
    long long stride = ((long long)gridDim.x * blockDim.x) >> 5;

    for (long long e = gw; e < E; e += stride) {
        long long s = (long long)ei[e];       // src row of edge_index
        long long d = (long long)ei[(long long)E + e];  // dst row
        float al = alpha_dst[d] + alpha_src[s];
        al = (al >= 0.f) ? al : NEG_SLOPE * al;
        float w = __expf(al);
        float2 h = *reinterpret_cast<const float2*>(msgs + s * OUT_F + lane * 2);
        unsafeAtomicAdd(out + d * OUT_F + lane * 2,     w * h.x);
        unsafeAtomicAdd(out + d * OUT_F + lane * 2 + 1, w * h.y);
        if (lane == 0) unsafeAtomicAdd(denom + d, w);
    }
}

// ---------------------------------------------------------------------------
// Kernel 5: normalize out /= max(denom, EPS)
// ---------------------------------------------------------------------------
__global__ __launch_bounds__(256)
void gat_norm(float* __restrict__ out, const float* __restrict__ denom, int N)
{
    int t = blockIdx.x * 256 + threadIdx.x;
    if (t >= N * OUT_F) return;
    float dn = denom[t >> 6];
    dn = (dn > EPS) ? dn : EPS;
    out[t] = out[t] / dn;
}

extern "C" void kernel_launch(void* const* d_in, const int* in_sizes, int n_in,
                              void* d_out, int out_size, void* d_ws, size_t ws_size,
                              hipStream_t stream)
{
    const float* x  = (const float*)d_in[0];   // [N, 128]
    const float* W  = (const float*)d_in[1];   // [64, 128]
    const float* a  = (const float*)d_in[2];   // [128]
    const int*   ei = (const int*)d_in[3];     // [2, E]

    const int N = in_sizes[0] / IN_F;   // 100000
    const int E = in_sizes[3] / 2;      // 1600000

    float* ws        = (float*)d_ws;
    float* msgs      = ws;                        // N*64 floats
    float* alpha_dst = ws + (size_t)N * OUT_F;    // N floats
    float* alpha_src = alpha_dst + N;             // N floats
    float* denom     = alpha_src + N;             // N floats
    float* out       = (float*)d_out;             // N*64 floats (doubles as agg)

    const int total = N * OUT_F;

    gat_gemm    <<<(N + 127) / 128,       256, 0, stream>>>(x, W, msgs, N);
    gat_alpha   <<<(N + 255) / 256,       256, 0, stream>>>(msgs, a, alpha_dst, alpha_src, N);
    gat_selfloop<<<(total + 255) / 256,   256, 0, stream>>>(msgs, alpha_dst, alpha_src, out, denom, N);
    gat_edges   <<<8192,                  256, 0, stream>>>(ei, msgs, alpha_dst, alpha_src, out, denom, E);
    gat_norm    <<<(total + 255) / 256,   256, 0, stream>>>(out, denom, N);
}